// ChannelInteractionAttention_50036368999160
// MI455X (gfx1250) — compile-verified
//
#include <hip/hip_runtime.h>
#include <hip/hip_bf16.h>

typedef __attribute__((ext_vector_type(2))) float v2f;
typedef __attribute__((ext_vector_type(4))) float v4f;
typedef __attribute__((ext_vector_type(8))) float v8f;

#define BDIM 32
#define CDIM 512
#define HID  64
#define FAN1 768
#define G4   128   // pooled descriptor length (window-4 pool of x_c)
#define HW   4096  // 64*64 spatial elements per (b,c)

// ---------------------------------------------------------------------------
// Kernel 1: spatial mean pool. One block per (b,c); 256 threads read 4096
// floats as float4 and tree-reduce. Bandwidth-bound: 256 MB read.
// ---------------------------------------------------------------------------
__global__ __launch_bounds__(256) void pool_mean_kernel(
    const float* __restrict__ x, float* __restrict__ xc)
{
    const int bc = blockIdx.x;                       // 0 .. 16383
    const v4f* p = (const v4f*)(x + (size_t)bc * HW);
    const int t = threadIdx.x;

    float s = 0.f;
#pragma unroll
    for (int i = 0; i < 4; ++i) {
        v4f v = p[t + i * 256];
        s += v.x + v.y + v.z + v.w;
    }
    // wave32 reduce
#pragma unroll
    for (int off = 16; off > 0; off >>= 1)
        s += __shfl_down(s, off, 32);

    __shared__ float red[8];
    const int lane = t & 31, wv = t >> 5;
    if (lane == 0) red[wv] = s;
    __syncthreads();
    if (t == 0) {
        float tot = 0.f;
#pragma unroll
        for (int i = 0; i < 8; ++i) tot += red[i];
        xc[bc] = tot * (1.0f / (float)HW);
    }
}

// ---------------------------------------------------------------------------
// Kernel 2: single workgroup (1024 threads, 32 waves).
//   Waves 0..7: WMMA f32 16x16x4 GEMMs:
//     A[32,64]  = x_c[32,512] @ fc1_w[:, :512]^T          (K=512)
//     Bv[32,64] = g4[32,128]  @ (fc1_w[:,512:640]+fc1_w[:,640:768])^T (K=128)
//   Then all 1024 threads: a[b,c] = sigmoid( sum_h relu(A+b1 + xc*Bv)*w2 + b2 )
// ---------------------------------------------------------------------------
__global__ __launch_bounds__(1024) void attn_weights_kernel(
    const float* __restrict__ fc1_w, const float* __restrict__ fc1_b,
    const float* __restrict__ fc2_w, const float* __restrict__ fc2_b,
    const float* __restrict__ xc,    float* __restrict__ a_out)
{
    __shared__ float g4s[BDIM][G4];     // 16 KB
    __shared__ float As [BDIM][HID];    //  8 KB (includes fc1_b)
    __shared__ float Bs [BDIM][HID];    //  8 KB
    __shared__ float w2s[HID];

    const int tid = threadIdx.x;

    // g4[b][q] = mean of 4 consecutive x_c values (x_c stays in global/L2)
    for (int i = tid; i < BDIM * G4; i += 1024) {
        const int b = i >> 7, q = i & (G4 - 1);
        const float* r = xc + b * CDIM + 4 * q;
        g4s[b][q] = 0.25f * (r[0] + r[1] + r[2] + r[3]);
    }
    if (tid < HID) w2s[tid] = fc2_w[tid];
    __syncthreads();

    const int wave = tid >> 5;
    const int lane = tid & 31;

    if (wave < 8) {
        // tile assignment: 2 M-tiles (b) x 4 N-tiles (h), one per wave
        const int m0 = (wave >> 2) * 16;
        const int n0 = (wave & 3) * 16;
        // V_WMMA_F32_16X16X4_F32 frag layout:
        //   A: vgpr0 = (M=lane%16, K=k0+{0 lo-half | 2 hi-half}), vgpr1 = K+1 / K+3
        //   B mirrored over K; C/D: vgpr r -> (M = r + 8*(lane>=16), N = lane%16)
        const int mrow = m0 + (lane & 15);
        const int ncol = n0 + (lane & 15);
        const int kofs = (lane < 16) ? 0 : 2;
        const float* wrow = fc1_w + (size_t)ncol * FAN1;

        v8f accA = {};
        for (int k = 0; k < CDIM; k += 4) {
            v2f af, bf;
            af.x = xc[mrow * CDIM + k + kofs];
            af.y = xc[mrow * CDIM + k + kofs + 1];
            bf.x = wrow[k + kofs];
            bf.y = wrow[k + kofs + 1];
            accA = __builtin_amdgcn_wmma_f32_16x16x4_f32(
                false, af, false, bf, (short)0, accA, false, false);
        }

        v8f accB = {};
        for (int k = 0; k < G4; k += 4) {
            v2f af, bf;
            af.x = g4s[mrow][k + kofs];
            af.y = g4s[mrow][k + kofs + 1];
            bf.x = wrow[512 + k + kofs]     + wrow[640 + k + kofs];
            bf.y = wrow[512 + k + kofs + 1] + wrow[640 + k + kofs + 1];
            accB = __builtin_amdgcn_wmma_f32_16x16x4_f32(
                false, af, false, bf, (short)0, accB, false, false);
        }

        const float b1 = fc1_b[ncol];
        const int rbase = m0 + ((lane >= 16) ? 8 : 0);
        const int col = n0 + (lane & 15);
#pragma unroll
        for (int r = 0; r < 8; ++r) {
            As[rbase + r][col] = accA[r] + b1;
            Bs[rbase + r][col] = accB[r];
        }
    }
    __syncthreads();

    // per-(b,c) attention scalar
    const float bias2 = fc2_b[0];
    for (int i = tid; i < BDIM * CDIM; i += 1024) {
        const int b = i >> 9;
        const float xcb = xc[i];
        float acc = bias2;
#pragma unroll 8
        for (int h = 0; h < HID; ++h) {
            float t = fmaf(xcb, Bs[b][h], As[b][h]);
            t = fmaxf(t, 0.f);
            acc = fmaf(t, w2s[h], acc);
        }
        a_out[i] = 1.0f / (1.0f + __expf(-acc));
    }
}

// ---------------------------------------------------------------------------
// Kernel 3: out[b,c,:,:] = x[b,c,:,:] * a[b,c]. Streaming: 256 MB in,
// 256 MB out. Blocks iterate in REVERSE so the tail of x (still resident in
// the 192 MB L2 after kernel 1) hits; output uses non-temporal stores since
// it is never re-read.
// ---------------------------------------------------------------------------
__global__ __launch_bounds__(256) void scale_kernel(
    const float* __restrict__ x, const float* __restrict__ a,
    float* __restrict__ out)
{
    const int bc = (int)gridDim.x - 1 - (int)blockIdx.x;
    const float s = a[bc];
    const v4f* xin = (const v4f*)(x + (size_t)bc * HW);
    v4f* o = (v4f*)(out + (size_t)bc * HW);
    const int t = threadIdx.x;
#pragma unroll
    for (int i = 0; i < 4; ++i) {
        v4f v = xin[t + i * 256];
        v *= s;
        __builtin_nontemporal_store(v, &o[t + i * 256]);
    }
}

// ---------------------------------------------------------------------------
extern "C" void kernel_launch(void* const* d_in, const int* in_sizes, int n_in,
                              void* d_out, int out_size, void* d_ws, size_t ws_size,
                              hipStream_t stream)
{
    const float* x     = (const float*)d_in[0];  // [32,512,64,64]
    const float* fc1_w = (const float*)d_in[1];  // [64,768]
    const float* fc1_b = (const float*)d_in[2];  // [64]
    const float* fc2_w = (const float*)d_in[3];  // [1,64]
    const float* fc2_b = (const float*)d_in[4];  // [1]
    float* out = (float*)d_out;                  // [32,512,64,64]

    float* xc = (float*)d_ws;                    // 16384 floats
    float* a  = xc + BDIM * CDIM;                // 16384 floats

    pool_mean_kernel<<<BDIM * CDIM, 256, 0, stream>>>(x, xc);
    attn_weights_kernel<<<1, 1024, 0, stream>>>(fc1_w, fc1_b, fc2_w, fc2_b, xc, a);
    scale_kernel<<<BDIM * CDIM, 256, 0, stream>>>(x, a, out);
}